// BoxEncoder_36189394436612
// MI455X (gfx1250) — compile-verified
//
#include <hip/hip_runtime.h>
#include <hip/hip_bf16.h>
#include <stdint.h>

typedef __attribute__((ext_vector_type(2)))  float        v2f;
typedef __attribute__((ext_vector_type(8)))  float        v8f;
typedef __attribute__((ext_vector_type(4)))  unsigned int v4u;
typedef __attribute__((ext_vector_type(8)))  int          v8i;
typedef __attribute__((ext_vector_type(4)))  int          v4i;

#define NPRIOR   8732
#define NCLS     81
#define NB       8
#define PRE      400
#define TOPK     200
#define NTILES   ((NB * NPRIOR) / 16)   /* 4366, exact */
#define MIN_SC   0.01f
#define MAX_OVL  0.45f

/* ------------------------------------------------------------------ */
/* Kernel 1: decode boxes + softmax (WMMA f32 16x16x4 row-sums).       */
/* One wave handles a tile of 16 flattened priors. 8 waves / block.    */
/* ------------------------------------------------------------------ */
__global__ __launch_bounds__(256) void decode_softmax_kernel(
    const float* __restrict__ locs,     // [B,P,4]
    const float* __restrict__ scores,   // [B,P,81]
    const float* __restrict__ priors,   // [P,4] cxcywh
    float* __restrict__ probsT,         // [B,80,P]
    float* __restrict__ boxes_ws)       // [B*P,4] xyxy
{
    __shared__ float tile[8][16][84];   // exp values, padded 81->84
    __shared__ float sums[8][16];

    const int tid  = threadIdx.x;
    const int lane = tid & 31;
    const int wid  = tid >> 5;
    const int tidx = blockIdx.x * 8 + wid;      // tile index (wave-uniform)
    const bool tvalid = (tidx < NTILES);
    const int g0 = tidx * 16;                   // first flattened prior of tile

    /* phase 1: per-prior max + exp into LDS (2 lanes per prior) */
    if (tvalid) {
        const int p = lane >> 1;      // 0..15 local prior
        const int h = lane & 1;       // class-strided half
        const long gp = (long)(g0 + p);
        const float* sp = scores + gp * NCLS;
        float pm = -1e30f;
        for (int c = h; c < NCLS; c += 2) pm = fmaxf(pm, sp[c]);
        float m = fmaxf(pm, __shfl_xor(pm, 1, 32));
        for (int c = h; c < NCLS; c += 2) tile[wid][p][c] = expf(sp[c] - m);
        if (h == 0) { tile[wid][p][81] = 0.f; tile[wid][p][82] = 0.f; tile[wid][p][83] = 0.f; }
    }
    __syncthreads();

    /* phase 2: row sums via V_WMMA_F32_16X16X4_F32 with B = ones.
       A layout (32-bit 16x4): lane L holds M=L&15, K = (L>>4)*2 + {0,1}. */
    if (tvalid) {
        const int am = lane & 15;
        const int ak = (lane >> 4) * 2;
        v8f acc = {};
        const v2f bones = {1.0f, 1.0f};
        for (int k0 = 0; k0 < 84; k0 += 4) {
            v2f a;
            a.x = tile[wid][am][k0 + ak];
            a.y = tile[wid][am][k0 + ak + 1];
            acc = __builtin_amdgcn_wmma_f32_16x16x4_f32(
                false, a, false, bones, (short)0, acc, false, false);
        }
        /* D layout: lane 0 col N=0 holds M=0..7 in acc[0..7]; lane 16 holds M=8..15 */
        union { v8f v; float f[8]; } u; u.v = acc;
        if (lane == 0)  for (int m = 0; m < 8; ++m) sums[wid][m]     = u.f[m];
        if (lane == 16) for (int m = 0; m < 8; ++m) sums[wid][m + 8] = u.f[m];
    }
    __syncthreads();

    /* phase 3: write probsT (skip background class 0) + decode boxes */
    if (tvalid) {
        for (int idx = lane; idx < 16 * 80; idx += 32) {
            const int p = idx / 80;
            const int c = idx - p * 80;               // class-1 index 0..79
            const int g = g0 + p;
            const int b  = g / NPRIOR;
            const int pp = g - b * NPRIOR;
            probsT[((size_t)b * 80 + c) * NPRIOR + pp] =
                tile[wid][p][c + 1] / sums[wid][p];
        }
        if (lane < 16) {
            const int g = g0 + lane;
            const int b  = g / NPRIOR;
            const int pp = g - b * NPRIOR;
            const float4 l  = *(const float4*)(locs + (size_t)g * 4);
            const float4 pr = *(const float4*)(priors + (size_t)pp * 4);
            const float cx = l.x * pr.z * 0.1f + pr.x;
            const float cy = l.y * pr.w * 0.1f + pr.y;
            const float w  = expf(l.z * 0.2f) * pr.z;
            const float h  = expf(l.w * 0.2f) * pr.w;
            float4 bx;
            bx.x = cx - 0.5f * w; bx.y = cy - 0.5f * h;
            bx.z = cx + 0.5f * w; bx.w = cy + 0.5f * h;
            *(float4*)(boxes_ws + (size_t)g * 4) = bx;
        }
    }
}

/* ------------------------------------------------------------------ */
/* Kernel 2: per-(image,class) TDM-staged top-400 selection + NMS.    */
/* gridDim.x = 640 blocks, 256 threads.                               */
/* ------------------------------------------------------------------ */
__global__ __launch_bounds__(256) void topk_nms_kernel(
    const float* __restrict__ probsT,     // [B,80,P]
    const float* __restrict__ boxes_ws,   // [B*P,4]
    float* __restrict__ cand_boxes,       // [B,80,PRE,4]
    float* __restrict__ cand_scores)      // [B,80,PRE]
{
    __shared__ __align__(16) unsigned char pool[NPRIOR * 4];  // scores, later IoU bitmask
    float*    s    = (float*)pool;
    unsigned* mask = (unsigned*)pool;     // 400*13 words = 20.8KB <= 34.9KB
    __shared__ float  selV[PRE];
    __shared__ int    selI[PRE];
    __shared__ __align__(16) float4 cbox[PRE];
    __shared__ float  wv[8];
    __shared__ int    wi[8];
    __shared__ unsigned supLDS[13];
    __shared__ int    bcastI;

    const int bc  = blockIdx.x;           // b*80 + (c-1)
    const int b   = bc / 80;
    const int tid = threadIdx.x;
    const int lane = tid & 31;
    const int wid  = tid >> 5;
    const float* src = probsT + (size_t)bc * NPRIOR;

    /* --- TDM: stage the 8732-float class row into LDS (one wave issues) --- */
    if (wid == 0) {
        const unsigned lds_lo = (unsigned)(uintptr_t)(void*)s;          // LDS offset
        const unsigned long long ga = (unsigned long long)(uintptr_t)src;
        /* D# group0: count=1 | lds_addr | global_addr[56:0] | type=2   */
        v4u g0 = { 1u,
                   lds_lo,
                   (unsigned)(ga & 0xFFFFFFFFull),
                   (unsigned)((ga >> 32) & 0x1FFFFFFull) | 0x80000000u };
        /* D# group1: data_size=4B; tensor 8732x1; tile 8732x1; stride 8732 */
        v8i g1 = { (int)0x20000u,
                   (int)((unsigned)NPRIOR << 16),   // tensor_dim0 lo
                   (int)(1u << 16),                 // tensor_dim0 hi=0 | tensor_dim1=1
                   (int)((unsigned)NPRIOR << 16),   // tensor_dim1 hi=0 | tile_dim0
                   1,                               // tile_dim1=1 | tile_dim2=0
                   NPRIOR,                          // tensor_dim0_stride lo
                   0, 0 };
        v4i g2 = { 0, 0, 0, 0 };
        v4i g3 = { 0, 0, 0, 0 };
        v8i g4 = { 0, 0, 0, 0, 0, 0, 0, 0 };
        __builtin_amdgcn_tensor_load_to_lds(g0, g1, g2, g3, g4, 0);
        __builtin_amdgcn_s_wait_tensorcnt(0);
    }
    __syncthreads();

    /* threshold: s = (v > MIN) ? v : -1 */
    for (int i = tid; i < NPRIOR; i += 256) {
        const float v = s[i];
        s[i] = (v > MIN_SC) ? v : -1.0f;
    }
    __syncthreads();

    /* --- sorted top-400 selection (cached per-thread local argmax) --- */
    float lv = -1e30f; int li = 0x7fffffff;
    int owner = -1;
    for (int k = 0; k < PRE; ++k) {
        if (k == 0 || owner == tid) {
            lv = -1e30f; li = 0x7fffffff;
            for (int i = tid; i < NPRIOR; i += 256) {
                const float v = s[i];
                if (v > lv) { lv = v; li = i; }
            }
        }
        float rv = lv; int ri = li;
        for (int off = 16; off; off >>= 1) {
            const float ov = __shfl_xor(rv, off, 32);
            const int   oi = __shfl_xor(ri, off, 32);
            if (ov > rv || (ov == rv && oi < ri)) { rv = ov; ri = oi; }
        }
        if (lane == 0) { wv[wid] = rv; wi[wid] = ri; }
        __syncthreads();
        if (tid == 0) {
            float bv = wv[0]; int bi = wi[0];
            for (int w = 1; w < 8; ++w) {
                const float ov = wv[w]; const int oi = wi[w];
                if (ov > bv || (ov == bv && oi < bi)) { bv = ov; bi = oi; }
            }
            selV[k] = bv; selI[k] = bi; bcastI = bi;
            s[bi] = -3.0f;                      // remove (below all real values)
        }
        __syncthreads();
        owner = bcastI & 255;                   // stripe owner = index % 256
    }

    /* gather candidate boxes */
    for (int k = tid; k < PRE; k += 256)
        cbox[k] = *(const float4*)(boxes_ws + ((size_t)b * NPRIOR + selI[k]) * 4);
    __syncthreads();

    /* IoU > 0.45 bitmask: row i, word w covers j in [32w,32w+32)  (aliases s[]) */
    for (int item = tid; item < PRE * 13; item += 256) {
        const int i = item / 13;
        const int w = item - i * 13;
        const float4 A = cbox[i];
        const float areaA = (A.z - A.x) * (A.w - A.y);
        unsigned word = 0;
        const int jbase = w * 32;
        for (int bit = 0; bit < 32; ++bit) {
            const int j = jbase + bit;
            if (j < PRE) {
                const float4 B = cbox[j];
                const float lx = fmaxf(A.x, B.x), ly = fmaxf(A.y, B.y);
                const float rx = fminf(A.z, B.z), ry = fminf(A.w, B.w);
                const float iw = fmaxf(rx - lx, 0.f), ih = fmaxf(ry - ly, 0.f);
                const float inter = iw * ih;
                const float areaB = (B.z - B.x) * (B.w - B.y);
                const float uni = fmaxf(areaA + areaB - inter, 1e-10f);
                if (inter > MAX_OVL * uni) word |= (1u << bit);
            }
        }
        mask[item] = word;
    }
    __syncthreads();

    /* greedy scan: wave 0, 13 lanes own 32 suppression bits each */
    if (wid == 0) {
        unsigned sup = 0;
        for (int i = 0; i < PRE; ++i) {
            const int wsel = i >> 5, bit = i & 31;
            const unsigned supw = (unsigned)__shfl((int)sup, wsel, 32);
            const bool act = (selV[i] > MIN_SC) && !((supw >> bit) & 1u);
            if (act) {
                if (lane < 13) sup |= mask[i * 13 + lane];
                if (lane == wsel) sup &= ~(1u << bit);   // keep self
            }
        }
        if (lane < 13) supLDS[lane] = sup;
    }
    __syncthreads();

    /* emit candidates: suppressed/invalid carry score -1 */
    float*  cs = cand_scores + (size_t)bc * PRE;
    float4* cb = (float4*)(cand_boxes + (size_t)bc * PRE * 4);
    for (int k = tid; k < PRE; k += 256) {
        const bool keep = (selV[k] > MIN_SC) &&
                          !((supLDS[k >> 5] >> (k & 31)) & 1u);
        cs[k] = keep ? selV[k] : -1.0f;
        cb[k] = cbox[k];
    }
}

/* ------------------------------------------------------------------ */
/* Kernel 3: per-image stable top-200 over 80*400 candidates.         */
/* ------------------------------------------------------------------ */
__global__ __launch_bounds__(256) void final_topk_kernel(
    const float* __restrict__ cand_scores,  // [B,80,PRE]
    const float* __restrict__ cand_boxes,   // [B,80,PRE,4]
    float* __restrict__ out_boxes,          // [B,200,4]
    int*   __restrict__ out_labels,         // [B,200]
    float* __restrict__ out_scores)         // [B,200]
{
    const int N = 80 * PRE;                 // 32000
    __shared__ unsigned removed[N / 32];    // 4 KB
    __shared__ float outV[TOPK];
    __shared__ int   outI[TOPK];
    __shared__ float wv[8];
    __shared__ int   wi[8];
    __shared__ int   bcastI;

    const int b   = blockIdx.x;
    const int tid = threadIdx.x;
    const int lane = tid & 31;
    const int wid  = tid >> 5;
    const float* src = cand_scores + (size_t)b * N;

    for (int i = tid; i < N / 32; i += 256) removed[i] = 0;
    __syncthreads();

    float lv = -1e30f; int li = 0x7fffffff;
    int owner = -1;
    for (int k = 0; k < TOPK; ++k) {
        if (k == 0 || owner == tid) {
            lv = -1e30f; li = 0x7fffffff;
            for (int i = tid; i < N; i += 256) {
                if (!((removed[i >> 5] >> (i & 31)) & 1u)) {
                    const float v = src[i];
                    if (v > lv) { lv = v; li = i; }
                }
            }
        }
        float rv = lv; int ri = li;
        for (int off = 16; off; off >>= 1) {
            const float ov = __shfl_xor(rv, off, 32);
            const int   oi = __shfl_xor(ri, off, 32);
            if (ov > rv || (ov == rv && oi < ri)) { rv = ov; ri = oi; }
        }
        if (lane == 0) { wv[wid] = rv; wi[wid] = ri; }
        __syncthreads();
        if (tid == 0) {
            float bv = wv[0]; int bi = wi[0];
            for (int w = 1; w < 8; ++w) {
                const float ov = wv[w]; const int oi = wi[w];
                if (ov > bv || (ov == bv && oi < bi)) { bv = ov; bi = oi; }
            }
            outV[k] = bv; outI[k] = bi; bcastI = bi;
            removed[bi >> 5] |= 1u << (bi & 31);
        }
        __syncthreads();
        owner = bcastI & 255;
    }

    for (int k = tid; k < TOPK; k += 256) {
        const int f = outI[k];
        const int c = f / PRE;
        const int slot = f - c * PRE;
        const float4 bx = *(const float4*)(
            cand_boxes + (((size_t)b * 80 + c) * PRE + slot) * 4);
        *(float4*)(out_boxes + ((size_t)b * TOPK + k) * 4) = bx;
        out_labels[b * TOPK + k] = c + 1;
        out_scores[b * TOPK + k] = outV[k];
    }
}

/* ------------------------------------------------------------------ */
extern "C" void kernel_launch(void* const* d_in, const int* in_sizes, int n_in,
                              void* d_out, int out_size, void* d_ws, size_t ws_size,
                              hipStream_t stream) {
    const float* locs   = (const float*)d_in[0];   // [8,8732,4]
    const float* scores = (const float*)d_in[1];   // [8,8732,81]
    const float* priors = (const float*)d_in[2];   // [8732,4]
    (void)in_sizes; (void)n_in; (void)out_size; (void)ws_size;

    /* workspace partition (floats): probsT | boxes | cand_boxes | cand_scores */
    float* ws          = (float*)d_ws;
    float* probsT      = ws;                                          // 8*80*8732
    float* boxes_ws    = probsT + (size_t)NB * 80 * NPRIOR;           // 8*8732*4
    float* cand_boxes  = boxes_ws + (size_t)NB * NPRIOR * 4;          // 8*80*400*4
    float* cand_scores = cand_boxes + (size_t)NB * 80 * PRE * 4;      // 8*80*400

    float* out        = (float*)d_out;
    float* out_boxes  = out;                                          // 8*200*4
    int*   out_labels = (int*)(out + (size_t)NB * TOPK * 4);          // 8*200
    float* out_scores = out + (size_t)NB * TOPK * 4 + (size_t)NB * TOPK;

    const int k1_blocks = (NTILES + 7) / 8;   // 546
    decode_softmax_kernel<<<k1_blocks, 256, 0, stream>>>(
        locs, scores, priors, probsT, boxes_ws);

    topk_nms_kernel<<<NB * 80, 256, 0, stream>>>(
        probsT, boxes_ws, cand_boxes, cand_scores);

    final_topk_kernel<<<NB, 256, 0, stream>>>(
        cand_scores, cand_boxes, out_boxes, out_labels, out_scores);
}